// GATConvGNN_14070312861758
// MI455X (gfx1250) — compile-verified
//
#include <hip/hip_runtime.h>
#include <hip/hip_bf16.h>
#include <math.h>

// ---------------------------------------------------------------------------
// GATConv GNN on MI455X (gfx1250): all GEMMs via V_WMMA_F32_16X16X4_F32,
// software-pipelined inner loop (prefetch next K-step before current WMMAs).
// ---------------------------------------------------------------------------

typedef __attribute__((ext_vector_type(2))) float v2f;
typedef __attribute__((ext_vector_type(8))) float v8f;

static constexpr int NN   = 4096;   // nodes
static constexpr int FIN  = 2048;   // input feature dim
static constexpr int HD   = 256;    // hidden dim per head
static constexpr int NCLS = 10;     // classes

__device__ __forceinline__ v8f wmma_f32(v2f a, v2f b, v8f c) {
    // 8-arg pattern: (neg_a, A, neg_b, B, c_mod, C, reuse_a, reuse_b)
    return __builtin_amdgcn_wmma_f32_16x16x4_f32(
        false, a, false, b, (short)0, c, false, false);
}

// ---------------------------------------------------------------------------
// Generic fp32 WMMA GEMM: C[M,Ncols] (+)= A[M,K] * B  (B is [K,Ncols], or
// [Ncols,K] read transposed when TRANSB). Block = 256 threads = 8 waves,
// block tile 128x128; each wave owns a 32x64 panel (2 M-tiles x 4 N-tiles).
// Out-of-range rows/cols are handled by clamped loads + guarded stores, so
// EXEC is all-ones at every v_wmma. The K-loop is software-pipelined: the
// next step's 6 fragments are fetched into fresh registers before the 8
// WMMAs of the current step issue, decoupling global-load latency from the
// matrix pipe.
// ---------------------------------------------------------------------------
template<bool TRANSB, bool ACCUM>
__global__ __launch_bounds__(256)
void gemm_wmma(const float* __restrict__ A, const float* __restrict__ B,
               float* __restrict__ C, int M, int Ncols, int K,
               int lda, int ldb, int ldc)
{
    const int lane = threadIdx.x & 31;
    const int wave = threadIdx.x >> 5;
    const int wm   = wave >> 1;              // 0..3
    const int wn   = wave & 1;               // 0..1
    const int m0   = blockIdx.y * 128 + wm * 32;
    const int n0   = blockIdx.x * 128 + wn * 64;
    const int l15  = lane & 15;
    const int kh   = (lane >> 4) << 1;       // K sub-offset: 0 or 2

    // A fragment row pointers (clamped so loads are always in-bounds)
    int ar0 = m0 + l15;       if (ar0 > M - 1) ar0 = M - 1;
    int ar1 = m0 + 16 + l15;  if (ar1 > M - 1) ar1 = M - 1;
    const float* Ap0 = A + (size_t)ar0 * lda + kh;
    const float* Ap1 = A + (size_t)ar1 * lda + kh;

    // B fragment column pointers (clamped; garbage columns never stored)
    const float* Bp[4];
    #pragma unroll
    for (int nt = 0; nt < 4; ++nt) {
        int c = n0 + nt * 16 + l15;
        if (c > Ncols - 1) c = Ncols - 1;
        Bp[nt] = TRANSB ? (B + (size_t)c * ldb + kh)
                        : (B + (size_t)kh * ldb + c);
    }

    v8f acc[2][4] = {};

    // ---- prologue: fragments for k0 = 0 ----
    v2f af0 = *(const v2f*)(Ap0);
    v2f af1 = *(const v2f*)(Ap1);
    v2f bf[4];
    #pragma unroll
    for (int nt = 0; nt < 4; ++nt) {
        if (TRANSB) {
            bf[nt] = *(const v2f*)(Bp[nt]);
        } else {
            const float* p = Bp[nt];
            v2f t; t.x = p[0]; t.y = p[ldb];
            bf[nt] = t;
        }
    }

    for (int k0 = 0; k0 < K; k0 += 4) {
        // ---- prefetch next K-step (clamped on final iteration) ----
        const int k1 = (k0 + 4 < K) ? (k0 + 4) : k0;
        v2f naf0 = *(const v2f*)(Ap0 + k1);
        v2f naf1 = *(const v2f*)(Ap1 + k1);
        v2f nbf[4];
        #pragma unroll
        for (int nt = 0; nt < 4; ++nt) {
            if (TRANSB) {
                nbf[nt] = *(const v2f*)(Bp[nt] + k1);
            } else {
                const float* p = Bp[nt] + (size_t)k1 * ldb;
                v2f t; t.x = p[0]; t.y = p[ldb];
                nbf[nt] = t;
            }
        }
        // ---- 8 WMMAs on current fragments ----
        #pragma unroll
        for (int nt = 0; nt < 4; ++nt) {
            acc[0][nt] = wmma_f32(af0, bf[nt], acc[0][nt]);
            acc[1][nt] = wmma_f32(af1, bf[nt], acc[1][nt]);
        }
        af0 = naf0; af1 = naf1;
        #pragma unroll
        for (int nt = 0; nt < 4; ++nt) bf[nt] = nbf[nt];
    }

    // C/D layout: VGPR r, lanes 0-15 -> row m0+r, col n0+lane;
    //             lanes 16-31 -> row m0+r+8, col n0+lane-16.
    const int roff = (lane >> 4) << 3;       // 0 or 8
    #pragma unroll
    for (int mt = 0; mt < 2; ++mt) {
        const int rbase = m0 + mt * 16 + roff;
        #pragma unroll
        for (int nt = 0; nt < 4; ++nt) {
            const int col = n0 + nt * 16 + l15;
            if (col < Ncols) {
                #pragma unroll
                for (int r = 0; r < 8; ++r) {
                    const int row = rbase + r;
                    if (row < M) {
                        float* cp = C + (size_t)row * ldc + col;
                        if (ACCUM) *cp += acc[mt][nt][r];
                        else       *cp  = acc[mt][nt][r];
                    }
                }
            }
        }
    }
}

// ---------------------------------------------------------------------------
// Row L2-normalize: xn[n,:] = x[n,:] / ||x[n,:]||   (one wave32 per row)
// ---------------------------------------------------------------------------
__global__ __launch_bounds__(256)
void rownorm_kernel(const float* __restrict__ x, float* __restrict__ xn)
{
    const int lane = threadIdx.x & 31;
    const int row  = blockIdx.x * 8 + (threadIdx.x >> 5);
    const float* xp = x + (size_t)row * FIN;
    float s = 0.f;
    for (int f = lane; f < FIN; f += 32) { float v = xp[f]; s += v * v; }
    #pragma unroll
    for (int o = 16; o; o >>= 1) s += __shfl_xor(s, o, 32);
    const float inv = 1.0f / sqrtf(s);     // exact-rounded; mask threshold is sensitive
    float* op = xn + (size_t)row * FIN;
    for (int f = lane; f < FIN; f += 32) op[f] = xp[f] * inv;
}

// maskb[i] = S[i] > 0.05
__global__ void mask_kernel(const float* __restrict__ S, unsigned char* __restrict__ maskb)
{
    size_t i = (size_t)blockIdx.x * blockDim.x + threadIdx.x;
    if (i < (size_t)NN * NN) maskb[i] = S[i] > 0.05f ? (unsigned char)1 : (unsigned char)0;
}

// el[n,h] = <feat[n,h,:], al[h,:]>, er likewise (one thread per (n,h))
__global__ void scores_kernel(const float* __restrict__ feat, const float* __restrict__ al,
                              const float* __restrict__ ar, float* __restrict__ el,
                              float* __restrict__ er, int H, int ldf)
{
    int i = blockIdx.x * blockDim.x + threadIdx.x;
    if (i >= NN * H) return;
    int n = i / H, h = i - n * H;
    const float* fp  = feat + (size_t)n * ldf + h * HD;
    const float* alp = al + h * HD;
    const float* arp = ar + h * HD;
    float sl = 0.f, sr = 0.f;
    for (int f = 0; f < HD; ++f) { float v = fp[f]; sl += v * alp[f]; sr += v * arp[f]; }
    el[i] = sl; er[i] = sr;
}

// Per-dst masked softmax stats over src: max and sum-exp, per head.
// One wave32 per dst row; self-loop guarantees a non-empty row.
__global__ __launch_bounds__(256)
void stats_kernel(const unsigned char* __restrict__ maskb, const float* __restrict__ el,
                  const float* __restrict__ er, float* __restrict__ mrow,
                  float* __restrict__ zrow, int H)
{
    const int lane = threadIdx.x & 31;
    const int dst  = blockIdx.x * 8 + (threadIdx.x >> 5);
    const unsigned char* mp = maskb + (size_t)dst * NN;
    for (int h = 0; h < H; ++h) {
        const float erd = er[dst * H + h];
        float mx = -1e30f;
        for (int s = lane; s < NN; s += 32) {
            if (mp[s]) {
                float e = erd + el[s * H + h];
                e = e > 0.f ? e : 0.2f * e;          // attn leaky-relu
                mx = fmaxf(mx, e);
            }
        }
        #pragma unroll
        for (int o = 16; o; o >>= 1) mx = fmaxf(mx, __shfl_xor(mx, o, 32));
        float sum = 0.f;
        for (int s = lane; s < NN; s += 32) {
            if (mp[s]) {
                float e = erd + el[s * H + h];
                e = e > 0.f ? e : 0.2f * e;
                sum += __expf(e - mx);
            }
        }
        #pragma unroll
        for (int o = 16; o; o >>= 1) sum += __shfl_xor(sum, o, 32);
        if (lane == 0) { mrow[dst * H + h] = mx; zrow[dst * H + h] = sum; }
    }
}

// W[dst,src] = mask ? exp(lrelu(er[dst]+el[src]) - m[dst]) / Z[dst] : 0
__global__ void alpha_kernel(const unsigned char* __restrict__ maskb, const float* __restrict__ el,
                             const float* __restrict__ er, const float* __restrict__ mrow,
                             const float* __restrict__ zrow, float* __restrict__ W,
                             int h, int H)
{
    size_t i = (size_t)blockIdx.x * blockDim.x + threadIdx.x;
    if (i >= (size_t)NN * NN) return;
    int d = (int)(i >> 12);
    int s = (int)(i & (NN - 1));
    float w = 0.f;
    if (maskb[i]) {
        float e = er[d * H + h] + el[s * H + h];
        e = e > 0.f ? e : 0.2f * e;
        w = __expf(e - mrow[d * H + h]) / zrow[d * H + h];
    }
    W[i] = w;
}

// out[n,f] = act( sum_h (agg[n,h,f] + res[n,h,f] + bias[h,f]) )
__global__ void headsum_kernel(const float* __restrict__ agg, const float* __restrict__ res,
                               const float* __restrict__ bias, float* __restrict__ out,
                               int H, int do_relu)
{
    int i = blockIdx.x * blockDim.x + threadIdx.x;
    if (i >= NN * HD) return;
    int n = i >> 8, f = i & (HD - 1);
    float s = 0.f;
    for (int h = 0; h < H; ++h) {
        int off = (n * H + h) * HD + f;
        s += agg[off] + res[off] + bias[h * HD + f];
    }
    out[i] = do_relu ? fmaxf(s, 0.f) : s;
}

// z[i] = leaky_relu(z[i] + b[i % 1024], 0.01)
__global__ void leaky_kernel(float* __restrict__ z, const float* __restrict__ b)
{
    int i = blockIdx.x * blockDim.x + threadIdx.x;
    if (i >= NN * 1024) return;
    float v = z[i] + b[i & 1023];
    z[i] = v > 0.f ? v : 0.01f * v;
}

// out[i] += b[i % NCLS]
__global__ void bias_kernel(float* __restrict__ out, const float* __restrict__ b)
{
    int i = blockIdx.x * blockDim.x + threadIdx.x;
    if (i >= NN * NCLS) return;
    out[i] += b[i % NCLS];
}

// ---------------------------------------------------------------------------
extern "C" void kernel_launch(void* const* d_in, const int* in_sizes, int n_in,
                              void* d_out, int out_size, void* d_ws, size_t ws_size,
                              hipStream_t stream)
{
    (void)in_sizes; (void)n_in; (void)out_size; (void)ws_size;
    const float* x   = (const float*)d_in[0];
    const float* W0  = (const float*)d_in[1];
    const float* al0 = (const float*)d_in[2];
    const float* ar0 = (const float*)d_in[3];
    const float* b0  = (const float*)d_in[4];
    const float* R0  = (const float*)d_in[5];
    const float* W1  = (const float*)d_in[6];
    const float* al1 = (const float*)d_in[7];
    const float* ar1 = (const float*)d_in[8];
    const float* b1  = (const float*)d_in[9];
    const float* R1  = (const float*)d_in[10];
    const float* W2  = (const float*)d_in[11];
    const float* al2 = (const float*)d_in[12];
    const float* ar2 = (const float*)d_in[13];
    const float* b2  = (const float*)d_in[14];
    const float* Wl1 = (const float*)d_in[15];
    const float* bl1 = (const float*)d_in[16];
    const float* Wc1 = (const float*)d_in[17];
    const float* bc1 = (const float*)d_in[18];
    const float* Wc2 = (const float*)d_in[19];
    const float* bc2 = (const float*)d_in[20];
    float* out = (float*)d_out;

    // Workspace layout (float offsets); regions reused once dead:
    //   [0 .. 8.4M)    xn, later feat (0..4.2M) + res (4.2M..8.4M)
    //   [8.4M..25.2M)  S / per-head alpha / later zpre
    float* ws   = (float*)d_ws;
    float* xn   = ws;
    float* feat = ws;
    float* res  = ws + 4194304;
    float* Sbuf = ws + 8388608;            // 16,777,216 floats
    float* agg  = ws + 25165824;           //  4,194,304 floats
    float* h1   = ws + 29360128;           //  1,048,576 floats (also h3)
    float* h2   = ws + 30408704;           //  1,048,576 floats
    float* el   = ws + 31457280;           //     16,384
    float* er   = ws + 31473664;
    float* mrow = ws + 31490048;
    float* zrow = ws + 31506432;
    unsigned char* maskb = (unsigned char*)(ws + 31522816);   // N*N bytes

    const dim3 blk(256);
    auto gg = [](int M, int Nc) { return dim3((unsigned)((Nc + 127) / 128),
                                              (unsigned)((M + 127) / 128)); };
    const size_t nn = (size_t)NN * NN;
    const unsigned gNN = (unsigned)((nn + 255) / 256);

    // ---- dynamic threshold graph ----
    rownorm_kernel<<<NN / 8, blk, 0, stream>>>(x, xn);
    gemm_wmma<true , false><<<gg(NN, NN), blk, 0, stream>>>(xn, xn, Sbuf,
                                                            NN, NN, FIN, FIN, FIN, NN);
    mask_kernel<<<gNN, blk, 0, stream>>>(Sbuf, maskb);

    // ---- GAT layer 1: 2048 -> 4x256, linear residual, relu ----
    gemm_wmma<false, false><<<gg(NN, 4 * HD), blk, 0, stream>>>(x, W0, feat,
                                                NN, 4 * HD, FIN, FIN, 4 * HD, 4 * HD);
    scores_kernel<<<(NN * 4 + 255) / 256, blk, 0, stream>>>(feat, al0, ar0, el, er, 4, 4 * HD);
    stats_kernel<<<NN / 8, blk, 0, stream>>>(maskb, el, er, mrow, zrow, 4);
    for (int h = 0; h < 4; ++h) {
        alpha_kernel<<<gNN, blk, 0, stream>>>(maskb, el, er, mrow, zrow, Sbuf, h, 4);
        gemm_wmma<false, false><<<gg(NN, HD), blk, 0, stream>>>(Sbuf, feat + h * HD,
                                    agg + h * HD, NN, HD, NN, NN, 4 * HD, 4 * HD);
    }
    gemm_wmma<false, false><<<gg(NN, 4 * HD), blk, 0, stream>>>(x, R0, res,
                                                NN, 4 * HD, FIN, FIN, 4 * HD, 4 * HD);
    headsum_kernel<<<(NN * HD + 255) / 256, blk, 0, stream>>>(agg, res, b0, h1, 4, 1);

    // ---- GAT layer 2: 256 -> 4x256, linear residual, relu ----
    gemm_wmma<false, false><<<gg(NN, 4 * HD), blk, 0, stream>>>(h1, W1, feat,
                                                NN, 4 * HD, HD, HD, 4 * HD, 4 * HD);
    scores_kernel<<<(NN * 4 + 255) / 256, blk, 0, stream>>>(feat, al1, ar1, el, er, 4, 4 * HD);
    stats_kernel<<<NN / 8, blk, 0, stream>>>(maskb, el, er, mrow, zrow, 4);
    for (int h = 0; h < 4; ++h) {
        alpha_kernel<<<gNN, blk, 0, stream>>>(maskb, el, er, mrow, zrow, Sbuf, h, 4);
        gemm_wmma<false, false><<<gg(NN, HD), blk, 0, stream>>>(Sbuf, feat + h * HD,
                                    agg + h * HD, NN, HD, NN, NN, 4 * HD, 4 * HD);
    }
    gemm_wmma<false, false><<<gg(NN, 4 * HD), blk, 0, stream>>>(h1, R1, res,
                                                NN, 4 * HD, HD, HD, 4 * HD, 4 * HD);
    headsum_kernel<<<(NN * HD + 255) / 256, blk, 0, stream>>>(agg, res, b1, h2, 4, 1);

    // ---- GAT layer 3: 1 head, identity residual, no relu ----
    gemm_wmma<false, false><<<gg(NN, HD), blk, 0, stream>>>(h2, W2, feat,
                                                NN, HD, HD, HD, HD, HD);
    scores_kernel<<<(NN + 255) / 256, blk, 0, stream>>>(feat, al2, ar2, el, er, 1, HD);
    stats_kernel<<<NN / 8, blk, 0, stream>>>(maskb, el, er, mrow, zrow, 1);
    alpha_kernel<<<gNN, blk, 0, stream>>>(maskb, el, er, mrow, zrow, Sbuf, 0, 1);
    gemm_wmma<false, false><<<gg(NN, HD), blk, 0, stream>>>(Sbuf, feat, agg,
                                                NN, HD, NN, NN, HD, HD);
    headsum_kernel<<<(NN * HD + 255) / 256, blk, 0, stream>>>(agg, h2, b2, h1, 1, 0);
    // h1 now holds final GAT output h (N x 256)

    // ---- MLP heads ----
    float* zpre = Sbuf;   // N x 1024 (alpha buffer dead)
    gemm_wmma<false, false><<<gg(NN, 1024), blk, 0, stream>>>(x, Wl1, zpre,
                                                NN, 1024, FIN, FIN, 1024, 1024);
    gemm_wmma<false, true ><<<gg(NN, 1024), blk, 0, stream>>>(h1, Wl1 + (size_t)FIN * 1024,
                                                zpre, NN, 1024, HD, HD, 1024, 1024);
    leaky_kernel<<<(NN * 1024 + 255) / 256, blk, 0, stream>>>(zpre, bl1);
    gemm_wmma<false, false><<<gg(NN, NCLS), blk, 0, stream>>>(zpre, Wc1, out,
                                                NN, NCLS, 1024, 1024, NCLS, NCLS);
    bias_kernel<<<(NN * NCLS + 255) / 256, blk, 0, stream>>>(out, bc1);
    gemm_wmma<false, false><<<gg(NN, NCLS), blk, 0, stream>>>(x, Wc2, out + NN * NCLS,
                                                NN, NCLS, FIN, FIN, NCLS, NCLS);
    bias_kernel<<<(NN * NCLS + 255) / 256, blk, 0, stream>>>(out + NN * NCLS, bc2);
}